// AlphaEntmaxHardAttention_47794396070418
// MI455X (gfx1250) — compile-verified
//
#include <hip/hip_runtime.h>
#include <hip/hip_bf16.h>
#include <math.h>

#define H   256
#define HQ  64           // H / 4 (float4 units)
#define NN  8192
#define GG  64
#define DEG 32
#define EE  (NN * (1 + DEG))
#define SS  33
#define S34 34

typedef __attribute__((ext_vector_type(2))) float v2f;
typedef __attribute__((ext_vector_type(8))) float v8f;

// ---------------- WMMA f32 GEMM: Out[16xH] = A[16xH] @ W[HxH] ----------------
// One wave per block; blockIdx.x selects the 16-wide N tile (16 tiles total).
// Uses V_WMMA_F32_16X16X4_F32 chained over K=256 (64 steps). A rows are always
// 16 (the 4-row static table is pre-padded with zeros), so A-fragment loads are
// unbranched 8-byte loads.
__global__ __launch_bounds__(32) void gemm16_wmma(const float* __restrict__ A,
                                                  const float* __restrict__ W,
                                                  float* __restrict__ Out) {
  const int lane = threadIdx.x & 31;
  const int n0   = blockIdx.x * 16;
  const int half = lane >> 4;   // 0: K pair {0,1}, 1: K pair {2,3}
  const int lm   = lane & 15;
  const int n    = n0 + lm;
  v8f acc = {};
  for (int k = 0; k < H; k += 4) {
    const int kb = k + half * 2;
    v2f a = *(const v2f*)(A + lm * H + kb);   // 8B-aligned: kb is even
    v2f b;
    b.x = W[kb * H + n];
    b.y = W[(kb + 1) * H + n];
    acc = __builtin_amdgcn_wmma_f32_16x16x4_f32(false, a, false, b, (short)0, acc,
                                                false, false);
  }
#pragma unroll
  for (int v = 0; v < 8; ++v) {
    const int row = v + 8 * half;
    Out[row * H + n0 + lm] = acc[v];
  }
}

// ---------------- pad emb_static (4xH) into zeroed 16xH staging --------------
__global__ void pad_static(const float* __restrict__ emb_static, float* __restrict__ P) {
  const int i = blockIdx.x * 256 + threadIdx.x;   // 16*256 total
  const int row = i >> 8;
  P[i] = (row < 4) ? emb_static[i] : 0.0f;
}

// ---------------- LayerNorm over 16 rows of a 16xH table (in place) ----------
__global__ __launch_bounds__(256) void ln_rows16(float* __restrict__ T,
                                                 const float* __restrict__ g,
                                                 const float* __restrict__ b) {
  __shared__ float red[256];
  const int row = blockIdx.x;
  const int tid = threadIdx.x;
  float x = T[row * H + tid];
  red[tid] = x;
  __syncthreads();
  for (int off = 128; off > 0; off >>= 1) {
    if (tid < off) red[tid] += red[tid + off];
    __syncthreads();
  }
  const float m = red[0] * (1.0f / H);
  __syncthreads();
  const float d = x - m;
  red[tid] = d * d;
  __syncthreads();
  for (int off = 128; off > 0; off >>= 1) {
    if (tid < off) red[tid] += red[tid + off];
    __syncthreads();
  }
  const float v = red[0] * (1.0f / H);
  T[row * H + tid] = d * rsqrtf(v + 1e-5f) * g[tid] + b[tid];
}

// ---------------- Gate scalar u for each of the 16 node-feature rows --------
__global__ __launch_bounds__(256) void gate_u(const float* __restrict__ G1,
                                              const float* __restrict__ bg1,
                                              const float* __restrict__ Wg2,
                                              const float* __restrict__ bg2,
                                              float* __restrict__ ut) {
  __shared__ float red[256];
  const int row = blockIdx.x;
  const int tid = threadIdx.x;
  float h = G1[row * H + tid] + bg1[tid];
  h = h > 0.0f ? h : 0.0f;
  red[tid] = h * Wg2[tid];
  __syncthreads();
  for (int off = 128; off > 0; off >>= 1) {
    if (tid < off) red[tid] += red[tid + off];
    __syncthreads();
  }
  if (tid == 0) ut[row] = 1.0f / (1.0f + expf(-(red[0] + bg2[0])));
}

// ---------------- workspace init ----------------
__global__ void init_ws(unsigned* __restrict__ nodemin, unsigned* __restrict__ edgemin,
                        int* __restrict__ slot2edge) {
  const int i = blockIdx.x * 256 + threadIdx.x;
  if (i < 1024) nodemin[i] = 0x7F800000u;          // +inf
  if (i < 16 * NN) edgemin[i] = 0x7F800000u;
  if (i < NN * SS) slot2edge[i] = -1;
}

// ---------------- node pass 1: state index + segmin --------------------------
__global__ void node_pass1(const int* __restrict__ ns, const float* __restrict__ scalars,
                           const int* __restrict__ batch, unsigned* __restrict__ nodemin,
                           int* __restrict__ nstate) {
  const int n = blockIdx.x * 256 + threadIdx.x;
  if (n >= NN) return;
  const int st = 2 * (ns[n * 3] + 2 * ns[n * 3 + 1] + 4 * ns[n * 3 + 2]);
  nstate[n] = st;
  atomicMin(&nodemin[st * GG + batch[n]], __float_as_uint(scalars[n]));
}

__global__ void node_pass2(const float* __restrict__ scalars, const int* __restrict__ batch,
                           const unsigned* __restrict__ nodemin, const int* __restrict__ nstate,
                           int* __restrict__ fidx) {
  const int n = blockIdx.x * 256 + threadIdx.x;
  if (n >= NN) return;
  const int st = nstate[n];
  const float mn = __uint_as_float(nodemin[st * GG + batch[n]]);
  fidx[n] = st + ((scalars[n] <= mn) ? 1 : 0);
}

// ---------------- edge pass 1: indices, static idx, segmin, slot scatter -----
__global__ void edge_pass1(const int* __restrict__ es, const float* __restrict__ scalars,
                           const int* __restrict__ src, const int* __restrict__ dst,
                           const int* __restrict__ slot, const int* __restrict__ nstate,
                           unsigned* __restrict__ edgemin, int* __restrict__ eidx,
                           int* __restrict__ sidx, int* __restrict__ slot2edge) {
  const int e = blockIdx.x * 256 + threadIdx.x;
  if (e >= EE) return;
  eidx[e] = es[e * 4] + 2 * es[e * 4 + 1] + 4 * es[e * 4 + 2] + 8 * es[e * 4 + 3];
  const float sc = scalars[e];
  const int d = dst[e], s = src[e];
  const float rs = scalars[d];   // receiver node scalar
  const float ss = scalars[s];   // sender node scalar
  const int rlx  = (sc < rs) ? 1 : 0;
  const int rlxd = ((ss + sc) < rs) ? 1 : 0;
  sidx[e] = rlx + 2 * rlxd;
  atomicMin(&edgemin[nstate[s] * NN + d], __float_as_uint(sc));
  slot2edge[d * SS + slot[e]] = e;
}

__global__ void edge_pass2(const float* __restrict__ scalars, const int* __restrict__ src,
                           const int* __restrict__ dst, const int* __restrict__ nstate,
                           const unsigned* __restrict__ edgemin, int* __restrict__ sbidx) {
  const int e = blockIdx.x * 256 + threadIdx.x;
  if (e >= EE) return;
  const int st = nstate[src[e]];
  const float mn = __uint_as_float(edgemin[st * NN + dst[e]]);
  sbidx[e] = st + ((scalars[e] <= mn) ? 1 : 0);
}

// ---------------- serial probability pipeline (lane 0, per node) -------------
__device__ void compute_attn(const float* __restrict__ z, const unsigned char* __restrict__ cm,
                             float u, int step, float* __restrict__ attn) {
  float zs[S34];
  for (int i = 0; i < S34; ++i) zs[i] = z[i];
  // descending insertion sort
  for (int i = 1; i < S34; ++i) {
    float key = zs[i];
    int j = i - 1;
    while (j >= 0 && zs[j] < key) { zs[j + 1] = zs[j]; --j; }
    zs[j + 1] = key;
  }
  float czarr[S34], tau_c[S34];
  float cz = 0.0f, cz2 = 0.0f;
  for (int i = 0; i < S34; ++i) {
    cz += zs[i];
    cz2 += zs[i] * zs[i];
    czarr[i] = cz;
    const float k = (float)(i + 1);
    const float mz = cz / k, mz2 = cz2 / k;
    float dd = mz * mz - mz2 + 1.0f / k;
    if (dd < 0.0f) dd = 0.0f;
    tau_c[i] = mz - sqrtf(dd + 1e-8f);
  }
  int kidx = 0;
  for (int i = 0; i < S34; ++i)
    if (zs[i] > tau_c[i]) ++kidx;
  const float tau15 = tau_c[kidx - 1];
  int ksp = 0;
  for (int i = 0; i < S34; ++i) {
    const float k = (float)(i + 1);
    if (k * zs[i] > czarr[i] - 1.0f) ++ksp;
  }
  const float tausp = (czarr[ksp - 1] - 1.0f) / (float)ksp;
  // softmax (max-subtracted, like jax.nn.softmax)
  const float m = zs[0];
  float soft[S34];
  float se = 0.0f;
  for (int i = 0; i < S34; ++i) { soft[i] = expf(z[i] - m); se += soft[i]; }
  for (int i = 0; i < S34; ++i) soft[i] /= se;
  // interpolated sparse probs
  float probs[S34];
  float hsum = 0.0f, spsum = 0.0f;
  for (int i = 0; i < S34; ++i) {
    float r15 = z[i] - tau15; if (r15 < 0.0f) r15 = 0.0f;
    const float p15 = r15 * r15;
    float rsp = z[i] - tausp; if (rsp < 0.0f) rsp = 0.0f;
    float p;
    if (u <= 0.5f) { const float w = u * 2.0f;          p = (1.0f - w) * soft[i] + w * p15; }
    else           { const float w = (u - 0.5f) * 2.0f; p = (1.0f - w) * p15 + w * rsp;    }
    probs[i] = p;
    if (p > 1e-4f) hsum += 1.0f;
    spsum += soft[i] * (cm[i] ? 1.0f : 0.0f);
  }
  // temperature / beta per reference
  float frac = (float)step * (1.0f / 1000.0f);
  if (frac < 0.0f) frac = 0.0f;
  if (frac > 1.0f) frac = 1.0f;
  float beta = 1.0f + (0.01f - 1.0f) * frac;
  if (beta < 0.01f) beta = 0.01f;
  if (beta > 1.0f) beta = 1.0f;
  for (int i = 0; i < S34; ++i) {
    const float hard = ((probs[i] > 1e-4f) ? 1.0f : 0.0f) / (hsum + 1e-9f);
    float a;
    if (step == -1) {
      a = hard;
    } else {
      float grad;
      if (beta > 0.01f) {
        const float sp = (soft[i] * (cm[i] ? 1.0f : 0.0f)) / (spsum + 1e-9f);
        grad = (1.0f - beta) * probs[i] + beta * sp;
      } else {
        grad = probs[i];
      }
      a = (hard - grad) + grad;   // stop_gradient forward value, matching fp order
    }
    attn[i] = a * (cm[i] ? 1.0f : 0.0f);
  }
}

// ---------------- attention: one wave per node -------------------------------
__global__ __launch_bounds__(32) void attn_kernel(
    const int* __restrict__ fidx, const int* __restrict__ slot2edge,
    const int* __restrict__ src_idx, const int* __restrict__ rev_idx,
    const int* __restrict__ eidx, const int* __restrict__ sidx, const int* __restrict__ sbidx,
    const float* __restrict__ Qt, const float* __restrict__ Kt, const float* __restrict__ Vt,
    const float* __restrict__ EKt, const float* __restrict__ EV1, const float* __restrict__ EV2,
    const float* __restrict__ EV3, const float* __restrict__ ut,
    const float* __restrict__ emb_virtual, const int* __restrict__ step_ptr,
    float* __restrict__ out1, float* __restrict__ agg) {
  __shared__ __align__(16) float Qs[H];
  __shared__ float logits[S34];
  __shared__ float attns[S34];
  __shared__ int vrow[S34], ei1[S34], ei2[S34], ei3[S34];
  __shared__ unsigned char valid[S34];
  const int n = blockIdx.x;
  const int lane = threadIdx.x;
  const int fi = fidx[n];
#pragma unroll
  for (int c = 0; c < 8; ++c) Qs[lane + 32 * c] = Qt[fi * H + lane + 32 * c];
  __syncthreads();
  const float4* Qv = (const float4*)Qs;
  for (int s = lane; s < S34; s += 32) {
    float lg = -1e9f;
    int vr = -1, a = -1, b = -1, c3 = -1, ok = 0;
    if (s == 0) {
      vr = fi;
      const float4* K1 = (const float4*)(Kt + fi * H);
      float acc = 0.0f;
      for (int hq = 0; hq < HQ; ++hq) {
        const float4 k = K1[hq];
        const float4 q = Qv[hq];
        acc += q.x * k.x + q.y * k.y + q.z * k.z + q.w * k.w;
      }
      lg = acc * (1.0f / 16.0f);
      ok = 1;
    } else {
      const int e = slot2edge[n * SS + (s - 1)];
      if (e >= 0) {
        const int sf = fidx[src_idx[e]];
        vr = sf;
        const float4* K1 = (const float4*)(Kt + sf * H);
        const float4* K2 = (const float4*)(EKt + sbidx[e] * H);
        a = eidx[e]; b = eidx[rev_idx[e]]; c3 = sidx[e];
        float acc = 0.0f;
        for (int hq = 0; hq < HQ; ++hq) {
          const float4 k1 = K1[hq];
          const float4 k2 = K2[hq];
          const float4 q = Qv[hq];
          acc += q.x * (k1.x + k2.x) + q.y * (k1.y + k2.y) +
                 q.z * (k1.z + k2.z) + q.w * (k1.w + k2.w);
        }
        lg = acc * (1.0f / 16.0f);
        ok = 1;
      }
    }
    logits[s] = lg;
    vrow[s] = vr; ei1[s] = a; ei2[s] = b; ei3[s] = c3;
    valid[s] = (unsigned char)ok;
  }
  __syncthreads();
  if (lane == 0) compute_attn(logits, valid, ut[fi], step_ptr[0], attns);
  __syncthreads();
  const float4* Vt4  = (const float4*)Vt;
  const float4* E1v  = (const float4*)EV1;
  const float4* E2v  = (const float4*)EV2;
  const float4* E3v  = (const float4*)EV3;
  const float4* EMBv = (const float4*)emb_virtual;
  float4* out1v = (float4*)out1;
  float4* aggv  = (float4*)agg;
#pragma unroll
  for (int c = 0; c < 2; ++c) {
    const int hq = lane + 32 * c;           // float4 column index in [0,64)
    float ax = 0.0f, ay = 0.0f, az = 0.0f, aw = 0.0f;
    for (int s = 0; s < S34; ++s) {
      if (!valid[s]) continue;
      float4 v = Vt4[vrow[s] * HQ + hq];
      if (s > 0) {
        const float4 e1 = E1v[ei1[s] * HQ + hq];
        const float4 e2 = E2v[ei2[s] * HQ + hq];
        const float4 e3 = E3v[ei3[s] * HQ + hq];
        v.x += e1.x + e2.x + e3.x;
        v.y += e1.y + e2.y + e3.y;
        v.z += e1.z + e2.z + e3.z;
        v.w += e1.w + e2.w + e3.w;
      }
      const float at = attns[s];
      ax += at * v.x; ay += at * v.y; az += at * v.z; aw += at * v.w;
    }
    const float4 nf = EMBv[fi * HQ + hq];
    float4 o;
    o.x = nf.x + ax; o.y = nf.y + ay; o.z = nf.z + az; o.w = nf.w + aw;
    out1v[n * HQ + hq] = o;
    float4 g;
    g.x = ax; g.y = ay; g.z = az; g.w = aw;
    aggv[n * HQ + hq] = g;
  }
}

// ---------------- streaming edge output (b128 loads/stores, coalesced) -------
__global__ void edge_out(const int* __restrict__ eidx, const int* __restrict__ dst,
                         const float4* __restrict__ emb_edge4, const float4* __restrict__ agg4,
                         float4* __restrict__ out2_4) {
  const int idx = blockIdx.x * 256 + threadIdx.x;   // E*HQ = 17,301,504 < 2^31
  const int e  = idx >> 6;                          // edge
  const int hq = idx & 63;                          // float4 column
  if (e >= EE) return;
  const float4 a = emb_edge4[eidx[e] * HQ + hq];
  const float4 b = agg4[dst[e] * HQ + hq];
  float4 o;
  o.x = a.x + b.x; o.y = a.y + b.y; o.z = a.z + b.z; o.w = a.w + b.w;
  out2_4[idx] = o;
}

// ---------------- workspace layout (float units) ----------------
#define OFF_QT    0
#define OFF_KT    4096
#define OFF_VT    8192
#define OFF_G1    12288
#define OFF_EKT   16384
#define OFF_C1    20480
#define OFF_C2    24576
#define OFF_C3    28672
#define OFF_EV1   32768
#define OFF_EV2   36864
#define OFF_EV3   40960
#define OFF_UT    45056
#define OFF_NMIN  45312
#define OFF_EMIN  46336
#define OFF_NST   177408
#define OFF_FIDX  185600
#define OFF_EIDX  193792
#define OFF_SIDX  464128
#define OFF_SBIDX 734464
#define OFF_S2E   1004800
#define OFF_AGG   1275136
#define OFF_PAD   3372288   // 16x256 zero-padded emb_static

extern "C" void kernel_launch(void* const* d_in, const int* in_sizes, int n_in,
                              void* d_out, int out_size, void* d_ws, size_t ws_size,
                              hipStream_t stream) {
  const int*   node_states = (const int*)d_in[0];
  const int*   edge_states = (const int*)d_in[1];
  const float* scalars     = (const float*)d_in[2];
  const int*   src_idx     = (const int*)d_in[3];
  const int*   dst_idx     = (const int*)d_in[4];
  const int*   slot_idx    = (const int*)d_in[5];
  const int*   rev_idx     = (const int*)d_in[6];
  const int*   batch_vec   = (const int*)d_in[7];
  const int*   step_ptr    = (const int*)d_in[8];
  const float* emb_edge    = (const float*)d_in[9];
  const float* emb_static  = (const float*)d_in[10];
  const float* emb_virtual = (const float*)d_in[11];
  const float* emb_reciev  = (const float*)d_in[12];
  const float* Wq   = (const float*)d_in[13];
  const float* Wk   = (const float*)d_in[14];
  const float* Wv   = (const float*)d_in[15];
  const float* Wek  = (const float*)d_in[16];
  const float* Wev  = (const float*)d_in[17];
  const float* Wcomb= (const float*)d_in[18];
  const float* gq = (const float*)d_in[19];
  const float* bq = (const float*)d_in[20];
  const float* gk = (const float*)d_in[21];
  const float* bk = (const float*)d_in[22];
  const float* gke= (const float*)d_in[23];
  const float* bke= (const float*)d_in[24];
  const float* Wg1= (const float*)d_in[25];
  const float* bg1= (const float*)d_in[26];
  const float* Wg2= (const float*)d_in[27];
  const float* bg2= (const float*)d_in[28];

  float* ws = (float*)d_ws;
  float* Qt  = ws + OFF_QT;   float* Kt  = ws + OFF_KT;  float* Vt  = ws + OFF_VT;
  float* G1  = ws + OFF_G1;   float* EKt = ws + OFF_EKT;
  float* C1  = ws + OFF_C1;   float* C2  = ws + OFF_C2;  float* C3  = ws + OFF_C3;
  float* EV1 = ws + OFF_EV1;  float* EV2 = ws + OFF_EV2; float* EV3 = ws + OFF_EV3;
  float* ut  = ws + OFF_UT;
  unsigned* nodemin = (unsigned*)(ws + OFF_NMIN);
  unsigned* edgemin = (unsigned*)(ws + OFF_EMIN);
  int* nstate = (int*)(ws + OFF_NST);
  int* fidx   = (int*)(ws + OFF_FIDX);
  int* eidx   = (int*)(ws + OFF_EIDX);
  int* sidx   = (int*)(ws + OFF_SIDX);
  int* sbidx  = (int*)(ws + OFF_SBIDX);
  int* s2e    = (int*)(ws + OFF_S2E);
  float* agg  = ws + OFF_AGG;
  float* Spad = ws + OFF_PAD;

  float* out1 = (float*)d_out;
  float* out2 = out1 + (size_t)NN * H;

  const int egrid = (EE + 255) / 256;   // 1056, also covers 16*NN and NN*SS inits

  init_ws<<<egrid, 256, 0, stream>>>(nodemin, edgemin, s2e);
  pad_static<<<16, 256, 0, stream>>>(emb_static, Spad);

  // Stage-1 WMMA table GEMMs
  gemm16_wmma<<<16, 32, 0, stream>>>(emb_virtual, Wq,  Qt);
  gemm16_wmma<<<16, 32, 0, stream>>>(emb_virtual, Wk,  Kt);
  gemm16_wmma<<<16, 32, 0, stream>>>(emb_virtual, Wv,  Vt);
  gemm16_wmma<<<16, 32, 0, stream>>>(emb_virtual, Wg1, G1);
  gemm16_wmma<<<16, 32, 0, stream>>>(emb_reciev,  Wek, EKt);
  gemm16_wmma<<<16, 32, 0, stream>>>(emb_edge,    Wcomb,             C1);
  gemm16_wmma<<<16, 32, 0, stream>>>(emb_edge,    Wcomb + 256 * 256, C2);
  gemm16_wmma<<<16, 32, 0, stream>>>(Spad,        Wcomb + 512 * 256, C3);
  // Stage-2 (dependent) WMMA GEMMs
  gemm16_wmma<<<16, 32, 0, stream>>>(C1, Wev, EV1);
  gemm16_wmma<<<16, 32, 0, stream>>>(C2, Wev, EV2);
  gemm16_wmma<<<16, 32, 0, stream>>>(C3, Wev, EV3);

  // LayerNorms + gate
  ln_rows16<<<16, 256, 0, stream>>>(Qt,  gq,  bq);
  ln_rows16<<<16, 256, 0, stream>>>(Kt,  gk,  bk);
  ln_rows16<<<16, 256, 0, stream>>>(EKt, gke, bke);
  gate_u<<<16, 256, 0, stream>>>(G1, bg1, Wg2, bg2, ut);

  // Graph index passes
  node_pass1<<<(NN + 255) / 256, 256, 0, stream>>>(node_states, scalars, batch_vec, nodemin, nstate);
  edge_pass1<<<egrid, 256, 0, stream>>>(edge_states, scalars, src_idx, dst_idx, slot_idx,
                                        nstate, edgemin, eidx, sidx, s2e);
  node_pass2<<<(NN + 255) / 256, 256, 0, stream>>>(scalars, batch_vec, nodemin, nstate, fidx);
  edge_pass2<<<egrid, 256, 0, stream>>>(scalars, src_idx, dst_idx, nstate, edgemin, sbidx);

  // Attention per node (one wave each)
  attn_kernel<<<NN, 32, 0, stream>>>(fidx, s2e, src_idx, rev_idx, eidx, sidx, sbidx,
                                     Qt, Kt, Vt, EKt, EV1, EV2, EV3, ut,
                                     emb_virtual, step_ptr, out1, agg);

  // Streaming edge outputs (dominant HBM traffic: ~277 MB write, b128 ops)
  edge_out<<<(EE * HQ) / 256, 256, 0, stream>>>(eidx, dst_idx, (const float4*)emb_edge,
                                                (const float4*)agg, (float4*)out2);
}